// AttentionUserEmbedding_30511447671145
// MI455X (gfx1250) — compile-verified
//
#include <hip/hip_runtime.h>
#include <hip/hip_bf16.h>
#include <math.h>

// ---------------------------------------------------------------------------
// AttentionUserEmbedding: out[b,d] = softmax_l(mask(E[b]·w))ᵀ · E[b]
// B=8192, L=200, D=64, f32.  HBM-bound (~419MB -> ~18us floor @23.3TB/s):
// single pass over E, staged per-row in LDS via global_load_async_to_lds_b128;
// scores via V_WMMA_F32_16X16X4_F32 with a column-broadcast B matrix.
// ---------------------------------------------------------------------------

typedef __attribute__((ext_vector_type(2))) float v2f;
typedef __attribute__((ext_vector_type(8))) float v8f;
typedef __attribute__((ext_vector_type(4))) int   v4i;

typedef __attribute__((address_space(1))) v4i* gptr_v4i;   // global int4*
typedef __attribute__((address_space(3))) v4i* lptr_v4i;   // LDS int4*

#define L_DIM 200
#define D_DIM 64
#define LROWS 208        // 13 tiles of 16 rows (rows 200..207 zeroed)
#define RSTRIDE 68       // floats per LDS row: 272B (16B aligned, bank-spread)
#define BDIM 256

#if __has_builtin(__builtin_amdgcn_global_load_async_to_lds_b128)
#define HAVE_ASYNC_LDS 1
#else
#define HAVE_ASYNC_LDS 0
#endif

__global__ __launch_bounds__(BDIM)
void attn_user_embed_kernel(const float* __restrict__ emb,
                            const int*   __restrict__ lengths,
                            const float* __restrict__ attn_w,
                            float*       __restrict__ out)
{
    __shared__ float s_emb[LROWS * RSTRIDE];   // 56,576 B
    __shared__ float s_w[D_DIM];
    __shared__ float s_scores[LROWS];          // scores, then exp-weights
    __shared__ float s_red[BDIM];
    __shared__ float s_part[4][D_DIM];

    const int tid  = threadIdx.x;
    const int b    = blockIdx.x;
    const int lane = tid & 31;
    const int wv   = tid >> 5;

    if (tid < D_DIM) s_w[tid] = attn_w[tid];

    // ---- Phase 1: stage E[b] (200x64 f32) into LDS, rows 200..207 = 0 ----
    const float* rowbase = emb + (size_t)b * (size_t)(L_DIM * D_DIM);
    for (int idx = tid; idx < LROWS * 16; idx += BDIM) {   // 3328 = 13*256
        const int l  = idx >> 4;
        const int c4 = (idx & 15) << 2;                    // float col
        float* dst = &s_emb[l * RSTRIDE + c4];
        if (l < L_DIM) {
            const float* src = rowbase + (size_t)l * D_DIM + c4;
#if HAVE_ASYNC_LDS
            __builtin_amdgcn_global_load_async_to_lds_b128(
                (gptr_v4i)(v4i*)src, (lptr_v4i)(v4i*)dst, 0, 0);
#else
            *(float4*)dst = *(const float4*)src;
#endif
        } else {
            float4 z = make_float4(0.f, 0.f, 0.f, 0.f);
            *(float4*)dst = z;
        }
    }
#if HAVE_ASYNC_LDS
#if __has_builtin(__builtin_amdgcn_s_wait_asynccnt)
    __builtin_amdgcn_s_wait_asynccnt(0);
#else
    asm volatile("s_wait_asynccnt 0" ::: "memory");
#endif
#endif
    __syncthreads();

    // ---- Phase 2: scores[l] = E[l,:]·w via WMMA f32 16x16x4 --------------
    // A lanes 0..15 hold K=4j+{0,1}, lanes 16..31 hold K=4j+{2,3} (ISA layout).
    // B[k,n] = w[4j+k] for ALL n  ->  D[m,n] = score(l0+m) in every column.
    {
        const int h = lane >> 4;       // 0 | 1
        const int m = lane & 15;       // row within tile
        for (int t = wv; t < 13; t += 8) {         // wave-uniform -> EXEC all 1s
            const int l0 = t << 4;
            const float* arow = &s_emb[(l0 + m) * RSTRIDE + 2 * h];
            v8f c = {};
#pragma unroll
            for (int j = 0; j < 16; ++j) {         // K = 64 in chunks of 4
                v2f a, bb;
                a.x  = arow[4 * j];
                a.y  = arow[4 * j + 1];
                bb.x = s_w[4 * j + 2 * h];
                bb.y = s_w[4 * j + 2 * h + 1];
                c = __builtin_amdgcn_wmma_f32_16x16x4_f32(
                        false, a, false, bb, (short)0, c, false, false);
            }
            if (m == 0) {                           // col 0 / col 16 writers
#pragma unroll
                for (int r = 0; r < 8; ++r)
                    s_scores[l0 + 8 * h + r] = c[r];
            }
        }
    }
    __syncthreads();

    // ---- Phase 3: masked softmax over l ----------------------------------
    const int len = lengths[b];
    float sc = (tid < L_DIM && tid < len) ? s_scores[tid] : -INFINITY;

    s_red[tid] = sc;
    __syncthreads();
#pragma unroll
    for (int off = BDIM / 2; off > 0; off >>= 1) {
        if (tid < off) s_red[tid] = fmaxf(s_red[tid], s_red[tid + off]);
        __syncthreads();
    }
    const float mx = s_red[0];
    __syncthreads();

    const float e = __expf(sc - mx);               // exp(-inf) = 0 for padding
    if (tid < LROWS) s_scores[tid] = e;            // unnormalized weights
    s_red[tid] = e;
    __syncthreads();
#pragma unroll
    for (int off = BDIM / 2; off > 0; off >>= 1) {
        if (tid < off) s_red[tid] += s_red[tid + off];
        __syncthreads();
    }
    const float inv_sum = 1.0f / s_red[0];
    __syncthreads();

    // ---- Phase 4: out[d] = sum_l w[l]*E[l,d] -----------------------------
    {
        const int d = tid & (D_DIM - 1);
        const int g = tid >> 6;                    // 4 l-partitions
        float acc = 0.f;
        for (int l = g; l < len; l += 4)
            acc += s_scores[l] * s_emb[l * RSTRIDE + d];   // broadcast + no-conflict
        s_part[g][d] = acc;
    }
    __syncthreads();
    if (tid < D_DIM) {
        const float r = (s_part[0][tid] + s_part[1][tid] +
                         s_part[2][tid] + s_part[3][tid]) * inv_sum;
        out[(size_t)b * D_DIM + tid] = r;
    }
}

extern "C" void kernel_launch(void* const* d_in, const int* in_sizes, int n_in,
                              void* d_out, int out_size, void* d_ws, size_t ws_size,
                              hipStream_t stream) {
    const float* emb     = (const float*)d_in[0];
    const int*   lengths = (const int*)  d_in[1];
    const float* attn_w  = (const float*)d_in[2];
    float*       out     = (float*)d_out;
    const int B = in_sizes[1];                     // lengths has B elements
    if (B <= 0) return;
    attn_user_embed_kernel<<<B, BDIM, 0, stream>>>(emb, lengths, attn_w, out);
}